// KGAT_model_23313082483398
// MI455X (gfx1250) — compile-verified
//
#include <hip/hip_runtime.h>

#define D       100
#define NV4     25        // 100 floats = 25 float4
#define N_USERS 50000
#define N_ITEMS 20000
#define N_ENT   50000
#define N_NODES (N_ITEMS + N_ENT)
#define K_NEI   20
#define NNZ     1000000
#define B_SZ    1024
#define N_PAIR  5120      // B * 5

typedef __attribute__((ext_vector_type(2))) float v2f;
typedef __attribute__((ext_vector_type(8))) float v8f;

// all_emb row r (r in [0, 70000)): first 20000 = news, rest = entity.
// Also serves as ent_base row (j in [0, 50000)).
__device__ __forceinline__ const float* all_emb_row(const float* news, const float* ent, int r) {
    return (r < N_ITEMS) ? (news + (size_t)r * D) : (ent + (size_t)(r - N_ITEMS) * D);
}

__global__ void k_zero_accs(float* accs) {
    accs[0] = 0.0f;   // rec_loss sum
    accs[1] = 0.0f;   // regularizer sum (||u||^2 + ||i||^2 over pairs)
}

// node_emb: one wave (32 lanes) per row; lanes 0..24 each own one float4 column.
__global__ void k_node_emb(const float* __restrict__ news, const float* __restrict__ ent,
                           const int* __restrict__ news_ent, const int* __restrict__ nei_ent,
                           float* __restrict__ node_emb) {
    int gid  = blockIdx.x * blockDim.x + threadIdx.x;
    int row  = gid >> 5;
    int lane = gid & 31;
    if (row >= N_NODES || lane >= NV4) return;

    const bool news_side = (row < N_ITEMS);
    const int* idxp;
    const float* base;
    if (news_side) {
        base = news + (size_t)row * D;
        idxp = news_ent + (size_t)row * K_NEI;
    } else {
        int j = row - N_ITEMS;
        base = all_emb_row(news, ent, j);      // ent_base[j]
        idxp = nei_ent + (size_t)j * K_NEI;
    }
    float4 acc = reinterpret_cast<const float4*>(base)[lane];
#pragma unroll 4
    for (int k = 0; k < K_NEI; ++k) {
        int j = idxp[k];
        const float* r = news_side ? (ent + (size_t)j * D)       // entity_emb gather
                                   : all_emb_row(news, ent, j);  // ent_base gather
        float4 x = reinterpret_cast<const float4*>(r)[lane];
        acc.x += x.x; acc.y += x.y; acc.z += x.z; acc.w += x.w;
    }
    reinterpret_cast<float4*>(node_emb + (size_t)row * D)[lane] = acc;
}

__global__ void k_copy4(const float4* __restrict__ src, float4* __restrict__ dst, int n4) {
    int i = blockIdx.x * blockDim.x + threadIdx.x;
    if (i < n4) dst[i] = src[i];
}

// segment_sum: user_acc[rows[e]] += vals[e] * node_emb[cols[e]]
// one thread per (edge, float4-column); 4 f32 atomics each.
__global__ void k_user_agg(const int* __restrict__ rows, const int* __restrict__ cols,
                           const float* __restrict__ vals,
                           const float* __restrict__ node_emb, float* __restrict__ user_acc) {
    int gid = blockIdx.x * blockDim.x + threadIdx.x;
    if (gid >= NNZ * NV4) return;
    int e = gid / NV4;
    int c = gid - e * NV4;
    int r  = rows[e];
    int ci = cols[e];
    float v = vals[e];
    float4 x = reinterpret_cast<const float4*>(node_emb + (size_t)ci * D)[c];
    float* dst = user_acc + (size_t)r * D + c * 4;
    atomicAdd(dst + 0, v * x.x);
    atomicAdd(dst + 1, v * x.y);
    atomicAdd(dst + 2, v * x.z);
    atomicAdd(dst + 3, v * x.w);
}

// in-place: node_emb[row] = all_emb[row] + 2 * node_emb[row] / max(||node_emb[row]||, eps)
__global__ void k_norm_node(const float* __restrict__ news, const float* __restrict__ ent,
                            float* __restrict__ node_emb) {
    int gid  = blockIdx.x * blockDim.x + threadIdx.x;
    int row  = gid >> 5;
    int lane = gid & 31;
    if (row >= N_NODES) return;
    float4 v = make_float4(0.f, 0.f, 0.f, 0.f);
    if (lane < NV4) v = reinterpret_cast<const float4*>(node_emb + (size_t)row * D)[lane];
    float ss = v.x * v.x + v.y * v.y + v.z * v.z + v.w * v.w;
#pragma unroll
    for (int off = 16; off; off >>= 1) ss += __shfl_xor(ss, off, 32);
    float s = 2.0f / fmaxf(sqrtf(ss), 1e-12f);
    if (lane < NV4) {
        float4 b = reinterpret_cast<const float4*>(all_emb_row(news, ent, row))[lane];
        float4 o = make_float4(b.x + v.x * s, b.y + v.y * s, b.z + v.z * s, b.w + v.w * s);
        reinterpret_cast<float4*>(node_emb + (size_t)row * D)[lane] = o;
    }
}

// in-place: user_acc[row] = user_emb_w[row] + 2 * user_acc[row] / max(||user_acc[row]||, eps)
__global__ void k_norm_user(const float* __restrict__ user_emb, float* __restrict__ user_acc) {
    int gid  = blockIdx.x * blockDim.x + threadIdx.x;
    int row  = gid >> 5;
    int lane = gid & 31;
    if (row >= N_USERS) return;
    float4 v = make_float4(0.f, 0.f, 0.f, 0.f);
    if (lane < NV4) v = reinterpret_cast<const float4*>(user_acc + (size_t)row * D)[lane];
    float ss = v.x * v.x + v.y * v.y + v.z * v.z + v.w * v.w;
#pragma unroll
    for (int off = 16; off; off >>= 1) ss += __shfl_xor(ss, off, 32);
    float s = 2.0f / fmaxf(sqrtf(ss), 1e-12f);
    if (lane < NV4) {
        float4 b = reinterpret_cast<const float4*>(user_emb + (size_t)row * D)[lane];
        float4 o = make_float4(b.x + v.x * s, b.y + v.y * s, b.z + v.z * s, b.w + v.w * s);
        reinterpret_cast<float4*>(user_acc + (size_t)row * D)[lane] = o;
    }
}

// WMMA scoring: one wave per 16 pairs. Gram tiles U*U^T, U*I^T, I*I^T via
// V_WMMA_F32_16X16X4_F32 over K=100 (25 steps of 4). Diagonals give
// scores (U*I^T) and regularizer terms (U*U^T, I*I^T).
// A fragment layout (16x4 f32): lane<16 holds {A[l][k0],A[l][k0+1]},
// lane>=16 holds {A[l-16][k0+2],A[l-16][k0+3]}; B (4x16) is structurally
// identical with N in place of M, so the same register pair serves both.
__global__ void k_score_wmma(const float* __restrict__ user_res, const float* __restrict__ node_res,
                             const int* __restrict__ user_index, const int* __restrict__ cand,
                             float* __restrict__ out_scores, float* __restrict__ accs) {
    int lane = threadIdx.x;
    int pairBase = blockIdx.x * 16;
    int l    = lane & 15;
    int half = lane >> 4;
    int pair = pairBase + l;
    const float* U = user_res + (size_t)user_index[pair / 5] * D;
    const float* I = node_res + (size_t)cand[pair] * D;

    v8f cUU = {}; v8f cUI = {}; v8f cII = {};
    for (int kk = 0; kk < NV4; ++kk) {
        int koff = kk * 4 + half * 2;
        float2 u2 = *reinterpret_cast<const float2*>(U + koff);
        float2 i2 = *reinterpret_cast<const float2*>(I + koff);
        v2f au = {u2.x, u2.y};
        v2f ai = {i2.x, i2.y};
        cUU = __builtin_amdgcn_wmma_f32_16x16x4_f32(false, au, false, au, (short)0, cUU, false, false);
        cUI = __builtin_amdgcn_wmma_f32_16x16x4_f32(false, au, false, ai, (short)0, cUI, false, false);
        cII = __builtin_amdgcn_wmma_f32_16x16x4_f32(false, ai, false, ai, (short)0, cII, false, false);
    }
    // C/D layout: VGPR r, lanes 0-15 -> (M=r, N=lane); lanes 16-31 -> (M=8+r, N=lane-16).
    // diag(m): m<8 at (vgpr m, lane m); m>=8 at (vgpr m-8, lane m+16).
    float reg = 0.0f;
    if (lane < 8) {
        out_scores[pairBase + lane] = cUI[lane];
        reg = cUU[lane] + cII[lane];
    } else if (lane >= 24) {
        int r = lane - 24;
        out_scores[pairBase + 8 + r] = cUI[r];
        reg = cUU[r] + cII[r];
    }
#pragma unroll
    for (int off = 16; off; off >>= 1) reg += __shfl_xor(reg, off, 32);
    if (lane == 0) atomicAdd(&accs[1], reg);
}

// per-row: probs = softmax(scores); logp = log_softmax(probs); -logp[argmax(labels)]
__global__ void k_loss(const float* __restrict__ scores, const float* __restrict__ labels,
                       float* __restrict__ accs) {
    int b = blockIdx.x * blockDim.x + threadIdx.x;
    if (b >= B_SZ) return;
    float s[5], p[5];
    float m = -3.402823466e+38f;
    for (int j = 0; j < 5; ++j) { s[j] = scores[b * 5 + j]; m = fmaxf(m, s[j]); }
    float se = 0.0f;
    for (int j = 0; j < 5; ++j) { p[j] = __expf(s[j] - m); se += p[j]; }
    float inv = 1.0f / se;
    float m2 = -3.402823466e+38f;
    for (int j = 0; j < 5; ++j) { p[j] *= inv; m2 = fmaxf(m2, p[j]); }
    float se2 = 0.0f;
    for (int j = 0; j < 5; ++j) se2 += __expf(p[j] - m2);
    float lse2 = __logf(se2) + m2;
    int t = 0; float best = labels[b * 5];
    for (int j = 1; j < 5; ++j) {
        float lv = labels[b * 5 + j];
        if (lv > best) { best = lv; t = j; }
    }
    atomicAdd(&accs[0], -(p[t] - lse2));
}

__global__ void k_final(const float* __restrict__ accs, float* __restrict__ out) {
    float rec = accs[0] / (float)B_SZ;
    float reg = accs[1] * 0.5f;
    float emb = 1e-5f * reg / (float)N_PAIR;
    out[0]          = rec + emb;
    out[1 + N_PAIR] = rec;
    out[2 + N_PAIR] = emb;
}

extern "C" void kernel_launch(void* const* d_in, const int* in_sizes, int n_in,
                              void* d_out, int out_size, void* d_ws, size_t ws_size,
                              hipStream_t stream) {
    const float* user_emb = (const float*)d_in[0];
    const float* news_emb = (const float*)d_in[1];
    const float* ent_emb  = (const float*)d_in[2];
    // d_in[3..7]: relation/attention params -> provably no effect (softmax over size-1 axis)
    const int*   news_ent   = (const int*)d_in[8];
    const int*   nei_ent    = (const int*)d_in[9];
    const int*   rows       = (const int*)d_in[11];
    const int*   cols       = (const int*)d_in[12];
    const float* vals       = (const float*)d_in[13];
    const int*   user_index = (const int*)d_in[14];
    const int*   cand       = (const int*)d_in[15];
    const float* labels     = (const float*)d_in[16];
    float* out = (float*)d_out;

    float* node_emb = (float*)d_ws;                            // 7,000,000 f32 (-> node_res in place)
    float* user_acc = node_emb + (size_t)N_NODES * D;          // 5,000,000 f32 (-> user_res in place)
    float* accs     = user_acc + (size_t)N_USERS * D;          // 2 f32

    k_zero_accs<<<1, 1, 0, stream>>>(accs);

    {   // node_emb (single hop; both hops identical)
        int thr = N_NODES * 32;
        k_node_emb<<<(thr + 255) / 256, 256, 0, stream>>>(news_emb, ent_emb, news_ent, nei_ent, node_emb);
    }
    {   // user_acc = user_emb_w
        int n4 = N_USERS * D / 4;
        k_copy4<<<(n4 + 255) / 256, 256, 0, stream>>>((const float4*)user_emb, (float4*)user_acc, n4);
    }
    {   // user_acc += A_sparse @ node_emb
        int thr = NNZ * NV4;
        k_user_agg<<<(thr + 255) / 256, 256, 0, stream>>>(rows, cols, vals, node_emb, user_acc);
    }
    {   // node_res / user_res (in place, includes the x2 from two identical hops)
        int thr = N_NODES * 32;
        k_norm_node<<<(thr + 255) / 256, 256, 0, stream>>>(news_emb, ent_emb, node_emb);
        thr = N_USERS * 32;
        k_norm_user<<<(thr + 255) / 256, 256, 0, stream>>>(user_emb, user_acc);
    }
    // scores + regularizer via WMMA (320 waves, 16 pairs each)
    k_score_wmma<<<N_PAIR / 16, 32, 0, stream>>>(user_acc, node_emb, user_index, cand, out + 1, accs);
    // double-softmax NLL
    k_loss<<<(B_SZ + 255) / 256, 256, 0, stream>>>(out + 1, labels, accs);
    k_final<<<1, 1, 0, stream>>>(accs, out);
}